// ManualGRU_64106681860391
// MI455X (gfx1250) — compile-verified
//
#include <hip/hip_runtime.h>
#include <hip/hip_bf16.h>

#define VOCABSZ 32000
#define HIDN    512
#define NGATE   1024              // 2*HID
#define BATCHN  64
#define SEQN    1024
#define NROWS   (BATCHN * SEQN)   // 65536
#define NTOT    1536              // gate cols (1024) + cand cols (512)
#define GRU_WGS 16

typedef __attribute__((ext_vector_type(2))) float v2f;
typedef __attribute__((ext_vector_type(8))) float v8f;

// ---- workspace layout (float offsets) -------------------------------------
static constexpr size_t XG_OFF  = 0;                                   // [SEQ][64][1024]
static constexpr size_t XG_SIZE = (size_t)SEQN * BATCHN * NGATE;       // 67,108,864 f
static constexpr size_t XC_OFF  = XG_OFF + XG_SIZE;                    // [SEQ][64][512]
static constexpr size_t XC_SIZE = (size_t)SEQN * BATCHN * HIDN;        // 33,554,432 f
static constexpr size_t H_OFF   = XC_OFF + XC_SIZE;                    // h  [64][512]
static constexpr size_t RH_OFF  = H_OFF + (size_t)BATCHN * HIDN;       // rh [64][512]
static constexpr size_t SY_OFF  = RH_OFF + (size_t)BATCHN * HIDN;      // 2 ints

// fp32 WMMA: D(16x16,f32) = A(16x4,f32) * B(4x16,f32) + C
static __device__ __forceinline__ v8f wmma_f32(v2f a, v2f b, v8f c) {
  return __builtin_amdgcn_wmma_f32_16x16x4_f32(false, a, false, b, (short)0, c,
                                               false, false);
}

// sense-reversing grid barrier over GRU_WGS resident workgroups
static __device__ __forceinline__ void grid_barrier(int* cnt, int* gen) {
  __threadfence();
  __syncthreads();
  if (threadIdx.x == 0) {
    int g = __hip_atomic_load(gen, __ATOMIC_RELAXED, __HIP_MEMORY_SCOPE_AGENT);
    int prev = __hip_atomic_fetch_add(cnt, 1, __ATOMIC_ACQ_REL,
                                      __HIP_MEMORY_SCOPE_AGENT);
    if (prev == GRU_WGS - 1) {
      __hip_atomic_store(cnt, 0, __ATOMIC_RELAXED, __HIP_MEMORY_SCOPE_AGENT);
      __hip_atomic_fetch_add(gen, 1, __ATOMIC_RELEASE, __HIP_MEMORY_SCOPE_AGENT);
    } else {
      while (__hip_atomic_load(gen, __ATOMIC_ACQUIRE,
                               __HIP_MEMORY_SCOPE_AGENT) == g)
        __builtin_amdgcn_s_sleep(2);
    }
  }
  __syncthreads();
}

// ---------------------------------------------------------------------------
// Kernel 1: fused embedding gather + GEMM producing gate_x / cand_x.
// Tile: 64(M) x 64(N), 128 threads = 4 waves; wave w owns N-fragment w.
// A (gathered embed rows) double-buffered: next K-chunk is gathered into
// VGPRs while the WMMA loop consumes the current LDS tile.
// Output stored t-major: row' = t*64 + b so the scan reads contiguous slabs.
// ---------------------------------------------------------------------------
__global__ void __launch_bounds__(128)
gru_pregemm(const long long* __restrict__ x, const float* __restrict__ embed,
            const float* __restrict__ W_gx, const float* __restrict__ b_gx,
            const float* __restrict__ W_cx, const float* __restrict__ b_cx,
            float* __restrict__ ws) {
  __shared__ int   tok[64];
  __shared__ float a_lds[64 * 66];   // pitch 66 to dodge bank conflicts

  const int tid  = threadIdx.x;
  const int wave = tid >> 5;
  const int lane = tid & 31;
  const int l15  = lane & 15;
  const int khalf = (lane >> 4) << 1;   // 0 or 2 (A/B K split across halves)
  const int rhi   = (lane >> 4) << 3;   // C/D row offset for upper half-wave
  const int m0 = blockIdx.y * 64;
  const int cb = blockIdx.x * 64;       // combined output col base in [0,1536)

  if (tid < 64) tok[tid] = (int)x[m0 + tid];

  const float* B;  int ldb, colb;  const float* bias;
  if (cb < NGATE) { B = W_gx; ldb = NGATE; colb = cb;         bias = b_gx; }
  else            { B = W_cx; ldb = HIDN;  colb = cb - NGATE; bias = b_cx; }

  const int ncol = colb + wave * 16 + l15;   // this lane's output column

  // staging registers: 2 rows per unrolled step (128 thr / 64-wide rows)
  const int rbase = tid >> 6;        // 0/1
  const int kkst  = tid & 63;
  float staged[32];

  __syncthreads();                   // tok ready
#pragma unroll
  for (int i = 0; i < 32; ++i)       // gather chunk kc=0
    staged[i] = embed[(size_t)tok[2 * i + rbase] * HIDN + kkst];

  v8f acc0 = {}, acc1 = {}, acc2 = {}, acc3 = {};

  for (int kc = 0; kc < HIDN; kc += 64) {
#pragma unroll
    for (int i = 0; i < 32; ++i)     // commit staged chunk to LDS
      a_lds[(2 * i + rbase) * 66 + kkst] = staged[i];
    if (kc + 64 < HIDN) {            // gather next chunk (overlaps WMMA loop)
#pragma unroll
      for (int i = 0; i < 32; ++i)
        staged[i] = embed[(size_t)tok[2 * i + rbase] * HIDN + kc + 64 + kkst];
    }
    __syncthreads();                 // a_lds ready

    const float* bp = B + (size_t)(kc + khalf) * ldb + ncol;
#pragma unroll 4
    for (int ks = 0; ks < 64; ks += 4) {
      v2f bf; bf.x = bp[0]; bf.y = bp[ldb];
      v2f a0 = *(const v2f*)&a_lds[(l15 +  0) * 66 + ks + khalf];
      v2f a1 = *(const v2f*)&a_lds[(l15 + 16) * 66 + ks + khalf];
      v2f a2 = *(const v2f*)&a_lds[(l15 + 32) * 66 + ks + khalf];
      v2f a3 = *(const v2f*)&a_lds[(l15 + 48) * 66 + ks + khalf];
      acc0 = wmma_f32(a0, bf, acc0);
      acc1 = wmma_f32(a1, bf, acc1);
      acc2 = wmma_f32(a2, bf, acc2);
      acc3 = wmma_f32(a3, bf, acc3);
      bp += 4 * (size_t)ldb;
    }
    __syncthreads();                 // consumers done before next overwrite
  }

  float bv = bias[ncol];
  float* XG = ws + XG_OFF;
  float* XC = ws + XC_OFF;
  auto store_frag = [&](v8f a, int mf) {
#pragma unroll
    for (int i = 0; i < 8; ++i) {
      int m = m0 + mf * 16 + i + rhi;         // original row = b*SEQ + t
      int b = m >> 10, t = m & 1023;
      float v = a[i] + bv;
      size_t orow = (size_t)(t * 64 + b);
      if (cb < NGATE) XG[orow * NGATE + ncol] = v;
      else            XC[orow * HIDN  + ncol] = v;
    }
  };
  store_frag(acc0, 0); store_frag(acc1, 1);
  store_frag(acc2, 2); store_frag(acc3, 3);
}

// ---------------------------------------------------------------------------
// Kernel 2: persistent GRU scan. 16 WGs x 256 threads (8 waves).
// WG k owns h columns [32k, 32k+32). Per step:
//   phase1: gates = sigmoid(XG_t + h@W_gh[:, z-cols | r-cols]); z -> LDS,
//           rh = r*h -> global.            <grid barrier>
//   phase2: h' = (1-z)*tanh(XC_t + rh@W_ch[:, cols]) + z*h.   <grid barrier>
// h/rh/weights stay L2-resident; XG/XC slabs (HBM for large t) are prefetched
// at phase start so the ~128-iter WMMA k-loop hides the HBM latency.
// ---------------------------------------------------------------------------
__global__ void __launch_bounds__(256)
gru_scan(const float* __restrict__ W_gh, const float* __restrict__ W_ch,
         float* __restrict__ ws, float* __restrict__ out) {
  __shared__ float z_lds[64 * 32];
  float* XG = ws + XG_OFF;
  float* XC = ws + XC_OFF;
  float* hg = ws + H_OFF;
  float* rh = ws + RH_OFF;
  int*   sy = (int*)(ws + SY_OFF);

  const int wg   = blockIdx.x;
  const int tid  = threadIdx.x;
  const int wave = tid >> 5;
  const int lane = tid & 31;
  const int l15  = lane & 15;
  const int khalf = (lane >> 4) << 1;
  const int rhi   = (lane >> 4) << 3;

  // phase-1 tiling: wave -> n-frag (0,1)=z-cols, (2,3)=r-cols; 2 m-frags each
  const int nf1 = wave & 3;
  const int mp1 = wave >> 2;                         // m-frags {2mp1, 2mp1+1}
  const int gcol = (nf1 < 2 ? (wg * 32 + nf1 * 16)
                            : (HIDN + wg * 32 + (nf1 - 2) * 16)) + l15;
  // phase-2 tiling: one 16x16 tile per wave
  const int mf2 = wave & 3;
  const int nf2 = wave >> 2;                         // 0 or 1
  const int hcol2 = wg * 32 + nf2 * 16 + l15;

  for (int t = 0; t < SEQN; ++t) {
    { // ---- phase 1 ----
      const float* xg = XG + (size_t)t * BATCHN * NGATE;
#pragma unroll
      for (int i = 0; i < 8; ++i) {                  // warm epilogue lines
        __builtin_prefetch(&xg[(mp1 * 32 + i + rhi) * NGATE + gcol], 0, 1);
        __builtin_prefetch(&xg[(mp1 * 32 + 16 + i + rhi) * NGATE + gcol], 0, 1);
      }
      v8f acc0 = {}, acc1 = {};
      const float* ap0 = hg + (size_t)(mp1 * 32 + l15) * HIDN + khalf;
      const float* ap1 = ap0 + 16 * HIDN;
      const float* bp  = W_gh + (size_t)khalf * NGATE + gcol;
#pragma unroll 4
      for (int k0 = 0; k0 < HIDN; k0 += 4) {
        v2f a0 = *(const v2f*)ap0;
        v2f a1 = *(const v2f*)ap1;
        v2f bf; bf.x = bp[0]; bf.y = bp[NGATE];
        acc0 = wmma_f32(a0, bf, acc0);
        acc1 = wmma_f32(a1, bf, acc1);
        ap0 += 4; ap1 += 4; bp += 4 * NGATE;
      }
#pragma unroll
      for (int i = 0; i < 8; ++i) {
        int r0 = mp1 * 32 + i + rhi;
        int r1 = r0 + 16;
        float v0 = acc0[i] + xg[r0 * NGATE + gcol];
        float v1 = acc1[i] + xg[r1 * NGATE + gcol];
        v0 = 1.f / (1.f + __expf(-v0));
        v1 = 1.f / (1.f + __expf(-v1));
        if (nf1 < 2) {                               // z gate -> LDS
          z_lds[r0 * 32 + (gcol - wg * 32)] = v0;
          z_lds[r1 * 32 + (gcol - wg * 32)] = v1;
        } else {                                     // r gate -> rh = r*h
          int hc = gcol - HIDN;
          rh[r0 * HIDN + hc] = v0 * hg[r0 * HIDN + hc];
          rh[r1 * HIDN + hc] = v1 * hg[r1 * HIDN + hc];
        }
      }
    }
    grid_barrier(sy, sy + 1);
    { // ---- phase 2 ----
      const float* xc = XC + (size_t)t * BATCHN * HIDN;
#pragma unroll
      for (int i = 0; i < 8; ++i)                    // warm epilogue lines
        __builtin_prefetch(&xc[(mf2 * 16 + i + rhi) * HIDN + hcol2], 0, 1);
      v8f acc = {};
      const float* ap = rh + (size_t)(mf2 * 16 + l15) * HIDN + khalf;
      const float* bp = W_ch + (size_t)khalf * HIDN + hcol2;
#pragma unroll 4
      for (int k0 = 0; k0 < HIDN; k0 += 4) {
        v2f a = *(const v2f*)ap;
        v2f bf; bf.x = bp[0]; bf.y = bp[HIDN];
        acc = wmma_f32(a, bf, acc);
        ap += 4; bp += 4 * HIDN;
      }
#pragma unroll
      for (int i = 0; i < 8; ++i) {
        int r = mf2 * 16 + i + rhi;
        float pre = acc[i] + xc[r * HIDN + hcol2];
        float e  = __expf(-2.f * __builtin_fabsf(pre));   // stable tanh
        float th = __builtin_copysignf((1.f - e) / (1.f + e), pre);
        float z  = z_lds[r * 32 + (hcol2 - wg * 32)];
        float ho = hg[r * HIDN + hcol2];
        float hn = (1.f - z) * th + z * ho;
        hg[r * HIDN + hcol2] = hn;
        if (t == SEQN - 1) out[r * HIDN + hcol2] = hn;
      }
    }
    grid_barrier(sy, sy + 1);
  }
}

__global__ void __launch_bounds__(256)
gru_init(float* __restrict__ ws) {
  int i = blockIdx.x * blockDim.x + threadIdx.x;
  if (i < BATCHN * HIDN) ws[H_OFF + i] = 0.f;
  if (i == 0) { int* s = (int*)(ws + SY_OFF); s[0] = 0; s[1] = 0; }
}

extern "C" void kernel_launch(void* const* d_in, const int* in_sizes, int n_in,
                              void* d_out, int out_size, void* d_ws,
                              size_t ws_size, hipStream_t stream) {
  (void)in_sizes; (void)n_in; (void)out_size; (void)ws_size;
  const long long* x  = (const long long*)d_in[0];
  const float* embed  = (const float*)d_in[1];
  const float* W_gx   = (const float*)d_in[2];
  const float* b_gx   = (const float*)d_in[3];
  const float* W_cx   = (const float*)d_in[4];
  const float* b_cx   = (const float*)d_in[5];
  const float* W_gh   = (const float*)d_in[6];
  const float* W_ch   = (const float*)d_in[7];
  float* out = (float*)d_out;
  float* ws  = (float*)d_ws;

  hipLaunchKernelGGL(gru_init, dim3((BATCHN * HIDN + 255) / 256), dim3(256), 0,
                     stream, ws);
  hipLaunchKernelGGL(gru_pregemm, dim3(NTOT / 64, NROWS / 64), dim3(128), 0,
                     stream, x, embed, W_gx, b_gx, W_cx, b_cx, ws);
  hipLaunchKernelGGL(gru_scan, dim3(GRU_WGS), dim3(256), 0, stream, W_gh, W_ch,
                     ws, out);
}